// CustomLoss_48430051230317
// MI455X (gfx1250) — compile-verified
//
#include <hip/hip_runtime.h>

typedef __attribute__((ext_vector_type(2))) float v2f;
typedef __attribute__((ext_vector_type(8))) float v8f;

#define NCOEF 18
#define NBLK  1024
#define NTHR  256

// Streaming loss kernel: one row per lane, grid-stride.
// Memory-bound (624 MB read-once): non-temporal loads, float2 on the
// 8B-aligned A rows. Wave reduction done with V_WMMA_F32_16X16X4_F32
// (A = per-lane partials, B = ones) -> deterministic, full f32 precision.
__global__ void loss_partial_kernel(const float* __restrict__ mo,   // [N,19] c||d
                                    const float* __restrict__ y,    // [N]
                                    const float* __restrict__ A,    // [N,18]
                                    const float* __restrict__ Btl,  // [N]
                                    const float* __restrict__ beta, // [18]
                                    float* __restrict__ partial,    // [NBLK]
                                    int n) {
    // Uniform beta -> SGPRs (compiler scalarizes: uniform pointer + index).
    float bet[NCOEF];
#pragma unroll
    for (int k = 0; k < NCOEF; ++k) bet[k] = beta[k];

    float acc = 0.0f;
    const int stride = gridDim.x * blockDim.x;
    for (int i = blockIdx.x * blockDim.x + threadIdx.x; i < n; i += stride) {
        const float* mrow = mo + (size_t)i * (NCOEF + 1);
        const float* arow = A  + (size_t)i * NCOEF;

        float t = __builtin_nontemporal_load(mrow + NCOEF); // d_i
#pragma unroll
        for (int k = 0; k < NCOEF; k += 2) {
            // A row base = 72*i bytes, k even -> 8-byte aligned float2.
            v2f av = __builtin_nontemporal_load((const v2f*)(arow + k));
            float c0 = __builtin_nontemporal_load(mrow + k);
            float c1 = __builtin_nontemporal_load(mrow + k + 1);
            t = fmaf(av.x, bet[k]     + c0, t);
            t = fmaf(av.y, bet[k + 1] + c1, t);
        }
        float e1 = t - __builtin_nontemporal_load(y + i);
        float e2 = t - __builtin_nontemporal_load(Btl + i);
        acc = fmaf(e1, e1, acc);
        acc = fmaf(e2, e2, acc);
    }

    // ---- Wave32 reduction via WMMA -------------------------------------
    // A (16x4 f32): lane l -> A[l%16, l<16?0:2] = acc, other K slots = 0.
    // B = ones(4x16)  =>  D[m,n] = acc_m + acc_{m+16} (same for all n).
    // Lanes 0-15 hold D rows 0-7 in v[0..7], lanes 16-31 hold rows 8-15.
    // EXEC is all-1s here (reconverged after the uniform-bound loop).
    v2f a; a.x = acc;  a.y = 0.0f;
    v2f b; b.x = 1.0f; b.y = 1.0f;
    v8f c8 = {};
    v8f d8 = __builtin_amdgcn_wmma_f32_16x16x4_f32(
        /*neg_a=*/false, a, /*neg_b=*/false, b,
        /*c_mod=*/(short)0, c8, /*reuse_a=*/false, /*reuse_b=*/false);

    float halfSum = d8[0] + d8[1] + d8[2] + d8[3] +
                    d8[4] + d8[5] + d8[6] + d8[7];
    // lanes 0-15: rows 0-7 sum; lanes 16-31: rows 8-15 sum -> combine.
    float waveSum = halfSum + __shfl_xor(halfSum, 16, 32);

    // ---- Block reduction (deterministic, no atomics) -------------------
    __shared__ float wsums[NTHR / 32];
    const int lane = threadIdx.x & 31;
    const int wid  = threadIdx.x >> 5;
    if (lane == 0) wsums[wid] = waveSum;
    __syncthreads();
    if (threadIdx.x == 0) {
        float s = 0.0f;
#pragma unroll
        for (int w = 0; w < NTHR / 32; ++w) s += wsums[w];
        partial[blockIdx.x] = s;
    }
}

// Final deterministic combine: fixed-order sum of NBLK partials.
__global__ void loss_final_kernel(const float* __restrict__ partial,
                                  float* __restrict__ out, int n) {
    if (blockIdx.x == 0 && threadIdx.x == 0) {
        double s = 0.0;
        for (int i = 0; i < NBLK; ++i) s += (double)partial[i];
        out[0] = (float)(s / (double)n);
    }
}

extern "C" void kernel_launch(void* const* d_in, const int* in_sizes, int n_in,
                              void* d_out, int out_size, void* d_ws, size_t ws_size,
                              hipStream_t stream) {
    const float* mo   = (const float*)d_in[0]; // model_output [N,19]
    const float* y    = (const float*)d_in[1]; // y   [N]
    const float* A    = (const float*)d_in[2]; // A   [N,18]
    const float* Btl  = (const float*)d_in[3]; // B_tl[N]
    const float* beta = (const float*)d_in[4]; // beta[18]
    const int n = in_sizes[1];                 // N rows

    float* partial = (float*)d_ws;             // NBLK floats of scratch

    loss_partial_kernel<<<NBLK, NTHR, 0, stream>>>(mo, y, A, Btl, beta, partial, n);
    loss_final_kernel<<<1, 32, 0, stream>>>(partial, (float*)d_out, n);
}